// Tree_Net_61203283968444
// MI455X (gfx1250) — compile-verified
//
#include <hip/hip_runtime.h>
#include <hip/hip_bf16.h>
#include <stdint.h>

typedef float v2f __attribute__((ext_vector_type(2)));
typedef float v8f __attribute__((ext_vector_type(8)));

#define BM 128
#define BN 128
#define BK 16
#define PAD 20  // LDS row pitch (floats): 80B rows keep 16B alignment for b128
                // async writes and make b64 fragment loads bank-conflict-free

#define TILE_F   (BM * PAD)            // floats per A (or B) tile
#define BUF_F    (2 * TILE_F)          // floats per double-buffer slot (A+B)

// Issue one 16B global -> LDS async transfer (CDNA5 Tensor/Async path, ASYNCcnt).
__device__ __forceinline__ void async_ld_b128(uint32_t lds_byte, const float* gaddr) {
    asm volatile("global_load_async_to_lds_b128 %0, %1, off"
                 :: "v"(lds_byte), "v"(gaddr) : "memory");
}
__device__ __forceinline__ void wait_async0() {
    asm volatile("s_wait_asynccnt 0x0" ::: "memory");
}

// C[M x N] = act( A[M x K] @ W^T (+ bias) ), W is (N x K) row-major.
// Dual-GEMM: for k >= khalf the weight source switches W1 -> W2
// (combined = X1@W1^T + X2@W2^T with A = seq_rep rows of length 2H).
__global__ __launch_bounds__(256) void gemm_f32_wmma(
    const float* __restrict__ A, int lda,
    const float* __restrict__ W1, const float* __restrict__ W2, int khalf, int ldw,
    const float* __restrict__ bias,
    float* __restrict__ C, int ldc,
    int M, int N, int K, int do_lrelu)
{
    __shared__ float smem[2 * BUF_F];   // [buf][A tile | B tile], 40 KB

    const int tid  = threadIdx.x;
    const int lane = tid & 31;
    const int wave = tid >> 5;
    const int wm   = wave & 3;     // 4 waves tile M (32 rows each)
    const int wn   = wave >> 2;    // 2 waves tile N (64 cols each)
    const int m0   = blockIdx.y * BM;
    const int n0   = blockIdx.x * BN;

    const int lrow = lane & 15;    // M (or N) row within a 16-tile
    const int lhi  = lane >> 4;    // 0 -> K+{0,1}, 1 -> K+{2,3} (ISA 16x4 f32 layout)

    // Low 32 bits of a flat shared pointer are the LDS byte address.
    const uint32_t smem_base = (uint32_t)(uintptr_t)(void*)&smem[0];

    // This thread's two staging slots (each covers one b128 = 4 floats).
    const int lin0 = tid,        r0 = lin0 >> 2, c0 = (lin0 & 3) << 2;
    const int lin1 = tid + 256,  r1 = lin1 >> 2, c1 = (lin1 & 3) << 2;

    // Stage A/B tiles for K-slab k0 into buffer `buf` via async-to-LDS.
    auto stage = [&](int k0, int buf) {
        const uint32_t base = smem_base + (uint32_t)(buf * BUF_F) * 4u;
        const float* Wsel = (k0 < khalf) ? W1 : W2;
        const int kc = (k0 < khalf) ? k0 : (k0 - khalf);
        async_ld_b128(base + (uint32_t)(r0 * PAD + c0) * 4u,
                      A + (size_t)(m0 + r0) * lda + k0 + c0);
        async_ld_b128(base + (uint32_t)(r1 * PAD + c1) * 4u,
                      A + (size_t)(m0 + r1) * lda + k0 + c1);
        // B rows beyond N are skipped (EXEC-masked): stale LDS only feeds
        // C columns >= N, which the store guard drops.
        if (n0 + r0 < N)
            async_ld_b128(base + (uint32_t)(TILE_F + r0 * PAD + c0) * 4u,
                          Wsel + (size_t)(n0 + r0) * ldw + kc + c0);
        if (n0 + r1 < N)
            async_ld_b128(base + (uint32_t)(TILE_F + r1 * PAD + c1) * 4u,
                          Wsel + (size_t)(n0 + r1) * ldw + kc + c1);
    };

    const v8f vzero = {0.f,0.f,0.f,0.f,0.f,0.f,0.f,0.f};
    v8f acc[2][4];
    for (int i = 0; i < 2; ++i)
        for (int j = 0; j < 4; ++j)
            acc[i][j] = vzero;

    stage(0, 0);                       // prologue: fill buffer 0
    int cur = 0;
    for (int k0 = 0; k0 < K; k0 += BK) {
        wait_async0();                 // this wave's tile for `cur` is in LDS
        __syncthreads();               // all waves' tiles ready; prev buf free
        if (k0 + BK < K) stage(k0 + BK, cur ^ 1);   // overlap next tile

        const int aoff = cur * BUF_F;
        const int boff = aoff + TILE_F;
        for (int kk = 0; kk < BK; kk += 4) {
            const int col = kk + 2 * lhi;
            v2f aF[2], bF[4];
            for (int i = 0; i < 2; ++i)
                aF[i] = *(const v2f*)(&smem[aoff + (wm * 32 + i * 16 + lrow) * PAD + col]);
            for (int j = 0; j < 4; ++j)
                bF[j] = *(const v2f*)(&smem[boff + (wn * 64 + j * 16 + lrow) * PAD + col]);
            for (int i = 0; i < 2; ++i)
                for (int j = 0; j < 4; ++j)
                    acc[i][j] = __builtin_amdgcn_wmma_f32_16x16x4_f32(
                        false, aF[i], false, bF[j], (short)0, acc[i][j],
                        false, false);
        }
        cur ^= 1;
    }

    // ---- epilogue: C/D layout -> VGPR r holds M = r + 8*lhi, N = lane&15 ----
    for (int i = 0; i < 2; ++i) {
        for (int j = 0; j < 4; ++j) {
            int nc = n0 + wn * 64 + j * 16 + lrow;
            if (nc >= N) continue;
            float bv = bias ? bias[nc] : 0.f;
            int mbase = m0 + wm * 32 + i * 16 + 8 * lhi;
            for (int r = 0; r < 8; ++r) {
                float v = acc[i][j][r] + bv;
                if (do_lrelu) v = (v > 0.f) ? v : 0.01f * v;
                C[(size_t)(mbase + r) * ldc + nc] = v;
            }
        }
    }
}

// One workgroup per batch chain. Circular correlation computed directly:
// c[k] = sum_j lv[j] * rv[(j+k) mod H]; the reference's pre-normalization of
// lv/rv cancels (bilinear op followed by output normalization).
__global__ __launch_bounds__(256) void tree_scan(
    float* __restrict__ leaves,   // combined, (B*S, H)
    float* __restrict__ phrase,   // (B*T, H)
    const int* __restrict__ info, // (B, T, 4): nc, parent, left, right
    int S_, int T_)
{
    constexpr int H = 512;
    __shared__ float lv[H];
    __shared__ float rv[H];
    __shared__ float wsum[8];
    __shared__ float snorm;

    const int b   = blockIdx.x;
    const int tid = threadIdx.x;

    for (int t = 0; t < T_; ++t) {
        const int4 ci = ((const int4*)info)[(size_t)b * T_ + t];
        const int ncount = ci.x, parent = ci.y, left = ci.z, right = ci.w;

        const float* ls = (left  < S_) ? leaves + ((size_t)b * S_ + left ) * H
                                       : phrase + ((size_t)b * T_ + left  - S_) * H;
        const float* rs = (right < S_) ? leaves + ((size_t)b * S_ + right) * H
                                       : phrase + ((size_t)b * T_ + right - S_) * H;
        lv[tid]       = ls[tid];
        lv[tid + 256] = ls[tid + 256];
        rv[tid]       = rs[tid];
        rv[tid + 256] = rs[tid + 256];
        __syncthreads();

        float a0, a1, inv = 1.f;
        if (ncount == 2) {             // uniform across block
            a0 = 0.f; a1 = 0.f;
            for (int j = 0; j < H; j += 4) {
                float4 av = *(const float4*)(&lv[j]);   // uniform b128 broadcast
                a0 = fmaf(av.x, rv[(j + 0 + tid) & (H - 1)], a0);
                a1 = fmaf(av.x, rv[(j + 0 + tid + 256) & (H - 1)], a1);
                a0 = fmaf(av.y, rv[(j + 1 + tid) & (H - 1)], a0);
                a1 = fmaf(av.y, rv[(j + 1 + tid + 256) & (H - 1)], a1);
                a0 = fmaf(av.z, rv[(j + 2 + tid) & (H - 1)], a0);
                a1 = fmaf(av.z, rv[(j + 2 + tid + 256) & (H - 1)], a1);
                a0 = fmaf(av.w, rv[(j + 3 + tid) & (H - 1)], a0);
                a1 = fmaf(av.w, rv[(j + 3 + tid + 256) & (H - 1)], a1);
            }
            float s = a0 * a0 + a1 * a1;
            for (int off = 16; off; off >>= 1) s += __shfl_xor(s, off, 32);
            if ((tid & 31) == 0) wsum[tid >> 5] = s;
            __syncthreads();
            if (tid == 0) {
                float tot = 0.f;
                for (int w = 0; w < 8; ++w) tot += wsum[w];
                snorm = rsqrtf(tot);
            }
            __syncthreads();
            inv = snorm;
        } else {
            a0 = lv[tid];
            a1 = lv[tid + 256];
        }

        if (ncount > 0) {
            float* d = (parent < S_) ? leaves + ((size_t)b * S_ + parent) * H
                                     : phrase + ((size_t)b * T_ + parent - S_) * H;
            d[tid]       = a0 * inv;
            d[tid + 256] = a1 * inv;
        }
        // Readers are in this workgroup (same WGP): commit stores to the
        // WGP-visible level, then barrier. Cheaper than a device fence.
        asm volatile("s_wait_storecnt 0x0" ::: "memory");
        __syncthreads();
    }
}

extern "C" void kernel_launch(void* const* d_in, const int* in_sizes, int n_in,
                              void* d_out, int out_size, void* d_ws, size_t ws_size,
                              hipStream_t stream)
{
    (void)in_sizes; (void)n_in; (void)out_size; (void)ws_size;
    constexpr int B = 256, S = 128, H = 512, CW = 511, CP = 800;
    constexpr int T = S - 1;

    const float* seq = (const float*)d_in[0];
    const float* W1  = (const float*)d_in[1];
    const float* W2  = (const float*)d_in[2];
    const float* Ww  = (const float*)d_in[3];
    const float* bw  = (const float*)d_in[4];
    const float* Wp  = (const float*)d_in[5];
    const float* bp  = (const float*)d_in[6];
    const int* cinfo = (const int*)d_in[7];
    const int* wlab  = (const int*)d_in[8];
    const int* plab  = (const int*)d_in[9];

    float* combined = (float*)d_ws;                    // (B*S, H) = 64 MB
    float* phrase   = combined + (size_t)B * S * H;    // (B*T, H) = 63.5 MB

    float* out_wo = (float*)d_out;                              // (B*S, CW)
    float* out_po = out_wo + (size_t)B * S * CW;                // (B*T, CP)
    char*  out_wl = (char*)(out_po + (size_t)B * T * CP);       // int32 labels
    char*  out_pl = out_wl + (size_t)B * S * sizeof(int);

    // 1) combined = leaky_relu(X1@W1^T + X2@W2^T)  (K=1024 with W1/W2 switch)
    gemm_f32_wmma<<<dim3(H / BN, (B * S) / BM), 256, 0, stream>>>(
        seq, 2 * H, W1, W2, /*khalf=*/H, /*ldw=*/H, nullptr,
        combined, H, B * S, H, 2 * H, /*lrelu=*/1);

    // 2) sequential tree composition (256 independent batch chains)
    tree_scan<<<B, 256, 0, stream>>>(combined, phrase, cinfo, S, T);

    // 3) word logits: combined @ Ww^T + bw  (N=511, tile-guarded)
    gemm_f32_wmma<<<dim3((CW + BN - 1) / BN, (B * S) / BM), 256, 0, stream>>>(
        combined, H, Ww, Ww, H, H, bw, out_wo, CW, B * S, CW, H, 0);

    // 4) phrase logits: phrase @ Wp^T + bp
    gemm_f32_wmma<<<dim3((CP + BN - 1) / BN, (B * T) / BM), 256, 0, stream>>>(
        phrase, H, Wp, Wp, H, H, bp, out_po, CP, B * T, CP, H, 0);

    // 5) label pass-through (int32 bit-copy, capture-safe D2D)
    hipMemcpyAsync(out_wl, wlab, (size_t)B * S * sizeof(int),
                   hipMemcpyDeviceToDevice, stream);
    hipMemcpyAsync(out_pl, plab, (size_t)B * T * sizeof(int),
                   hipMemcpyDeviceToDevice, stream);
}